// MultiLatentDecoderLayer_7576322310198
// MI455X (gfx1250) — compile-verified
//
#include <hip/hip_runtime.h>

typedef float v2f __attribute__((ext_vector_type(2)));
typedef float v8f __attribute__((ext_vector_type(8)));

static constexpr int NTOT = 131072;
static constexpr int D    = 16;
static constexpr int DIN  = 64;
static constexpr int DOUT = 64;
static constexpr int DC   = 4;                  // decoders staged per LDS chunk
static constexpr int QC   = DC * (DIN / 4);     // 64 k-steps (of K=4) per chunk
static constexpr int LDS_DW = QC * 4 * 64;      // 16384 dwords = 64 KB

// out = (A x B) * rowsum(alpha), where A[n,k] = alpha[k/64,n]*x[n,k%64],
// B[k,o] = scale[k/64][k%64][o].  One wave = 16 rows x 64 cols (4 C tiles).
__global__ __launch_bounds__(256)
void mld_wmma_kernel(const float* __restrict__ x,
                     const float* __restrict__ alpha,
                     const float* __restrict__ scale,
                     float* __restrict__ out) {
  __shared__ float ldsB[LDS_DW];

  const int tid  = threadIdx.x;
  const int lane = tid & 31;
  const int wave = tid >> 5;
  const int row  = lane & 15;
  const int hi   = lane >> 4;        // 0: lanes 0-15, 1: lanes 16-31
  const int c0   = hi << 1;          // K offset (0 or 2) inside a 4-wide k-step

  const int n0 = blockIdx.x * 128 + wave * 16;   // this wave's 16-row tile
  const int n  = n0 + row;

  // Stage this lane's A-side x data: for k-step t it needs x[n][4t+c0 .. 4t+c0+1]
  // (A layout: lane holds row l%16; VGPR0 = K0|K2, VGPR1 = K1|K3).
  v2f xr[16];
#pragma unroll
  for (int t = 0; t < 16; ++t)
    xr[t] = *(const v2f*)(x + (size_t)n * DIN + 4 * t + c0);

  v8f acc0 = {}, acc1 = {}, acc2 = {}, acc3 = {};
  float asum = 0.0f;

  for (int chunk = 0; chunk < D / DC; ++chunk) {
    const int dbase = chunk * DC;

    // ---- stage pre-swizzled B fragments for DC decoders into LDS ----
    // ldsB[((q*4+ob)*64) + 2c + {0,1}] = { S[d][i][o], S[d][i+1][o] }
    // with d = dbase + q/16, i = 4*(q%16) + 2*(c>=16), o = ob*16 + (c&15).
    // Inner loop then does one conflict-free ds_load_b64 per B fragment.
    for (int p = tid; p < QC * 4 * 32; p += 256) {
      const int c  = p & 31;
      const int ob = (p >> 5) & 3;
      const int q  = p >> 7;
      const int d  = dbase + (q >> 4);
      const int i  = ((q & 15) << 2) + ((c >> 4) << 1);
      const int o  = (ob << 4) + (c & 15);
      const float s0 = scale[(size_t)(d * DIN + i)     * DOUT + o];
      const float s1 = scale[(size_t)(d * DIN + i + 1) * DOUT + o];
      v2f pk; pk.x = s0; pk.y = s1;
      *(v2f*)(ldsB + (((q << 2) + ob) << 6) + (c << 1)) = pk;
    }
    __syncthreads();

    // ---- 64 WMMA k-steps over this chunk (4 decoders x 16 steps) ----
    for (int dd = 0; dd < DC; ++dd) {
      const int d   = dbase + dd;
      const float a = alpha[(size_t)d * NTOT + n];  // per-lane scalar (row n)
      asum += a;
      const int qb = dd << 4;
#pragma unroll
      for (int t = 0; t < 16; ++t) {
        v2f afrag;
        afrag.x = a * xr[t].x;     // A[n,k] = alpha*x; folds per-row weight into A
        afrag.y = a * xr[t].y;
        const float* bp = ldsB + (((qb + t) << 2) << 6) + (lane << 1);
        v2f b0 = *(const v2f*)(bp);
        v2f b1 = *(const v2f*)(bp + 64);
        v2f b2 = *(const v2f*)(bp + 128);
        v2f b3 = *(const v2f*)(bp + 192);
        acc0 = __builtin_amdgcn_wmma_f32_16x16x4_f32(false, afrag, false, b0,
                                                     (short)0, acc0, false, false);
        acc1 = __builtin_amdgcn_wmma_f32_16x16x4_f32(false, afrag, false, b1,
                                                     (short)0, acc1, false, false);
        acc2 = __builtin_amdgcn_wmma_f32_16x16x4_f32(false, afrag, false, b2,
                                                     (short)0, acc2, false, false);
        acc3 = __builtin_amdgcn_wmma_f32_16x16x4_f32(false, afrag, false, b3,
                                                     (short)0, acc3, false, false);
      }
    }
    __syncthreads();
  }

  // Broadcast per-row alpha-sum into the C-tile layout:
  // C VGPR r holds rows (r + 8*hi); asum for row m lives in lane m.
  float rs[8];
#pragma unroll
  for (int r = 0; r < 8; ++r) {
    const int srcByte = (r + (hi << 3)) << 2;
    rs[r] = __int_as_float(
        __builtin_amdgcn_ds_bpermute(srcByte, __float_as_int(asum)));
  }

  // w_out = asum * (A x B); store 16x64 tile (coalesced 64B per half-wave row)
#pragma unroll
  for (int r = 0; r < 8; ++r) {
    float* orow = out + (size_t)(n0 + r + (hi << 3)) * DOUT + (lane & 15);
    orow[0]  = rs[r] * acc0[r];
    orow[16] = rs[r] * acc1[r];
    orow[32] = rs[r] * acc2[r];
    orow[48] = rs[r] * acc3[r];
  }
}

extern "C" void kernel_launch(void* const* d_in, const int* in_sizes, int n_in,
                              void* d_out, int out_size, void* d_ws, size_t ws_size,
                              hipStream_t stream) {
  const float* x     = (const float*)d_in[0];
  const float* alpha = (const float*)d_in[1];
  const float* scale = (const float*)d_in[2];
  float* out = (float*)d_out;

  dim3 grid(NTOT / 128);   // 1024 blocks, 8 waves (16-row tiles) each
  dim3 block(256);
  hipLaunchKernelGGL(mld_wmma_kernel, grid, block, 0, stream,
                     x, alpha, scale, out);
}